// TransferNet_1640677507149
// MI455X (gfx1250) — compile-verified
//
#include <hip/hip_runtime.h>
#include <hip/hip_bf16.h>

// ---------------------------------------------------------------------------
// TransferNet clustering + classifier for MI455X (gfx1250, wave32, WMMA).
//
// Algebraic reduction: segment-summed cosine distances collapse to
//   argmin_c dist[b,c]  ==  argmax_c  feat[b] . (S[c]/count[c])
// where S[c] is the sum of L2-normalized bank rows of class c. This removes
// the 2048x65536 (512 MB) distance matrix entirely; the kernel is then
// HBM-bound on ~80 MB of reads (~3.5us at 23.3 TB/s).
// Both GEMMs use V_WMMA_F32_16X16X4_F32 (f32 in / f32 accum, matches ref).
// ---------------------------------------------------------------------------

typedef __attribute__((ext_vector_type(2))) float v2f;
typedef __attribute__((ext_vector_type(8))) float v8f;

#define BROWS     2048
#define DIN       2048
#define DB        256
#define NHIST     63488
#define NBANK     (BROWS + NHIST)   // 65536
#define NCLS      31
#define NCLSP     32
#define EPSN      1e-12f

// workspace layout (float units)
#define FEAT_OFF  0u                      // 2048*256
#define S_OFF     524288u                 // 32*256
#define CNT_OFF   (S_OFF + 8192u)         // 32
#define SCT_OFF   (CNT_OFF + 32u)         // 256*32  (S[c]/count[c], transposed)
#define WCP_OFF   (SCT_OFF + 8192u)       // 256*32  (Wc padded to 32 cols)
#define PART_OFF  (WCP_OFF + 8192u)       // 16 per-block loss partials

__device__ __forceinline__ v8f wmma4(v2f a, v2f b, v8f c) {
  // D = A(16x4 f32) * B(4x16 f32) + C(16x16 f32)
  return __builtin_amdgcn_wmma_f32_16x16x4_f32(
      /*neg_a=*/false, a, /*neg_b=*/false, b,
      /*c_mod=*/(short)0, c, /*reuse_a=*/false, /*reuse_b=*/false);
}

// ---------------------------------------------------------------------------
__global__ void zero_kernel(float* p, int n) {
  int i = blockIdx.x * blockDim.x + threadIdx.x;
  if (i < n) p[i] = 0.0f;
}

// ---------------------------------------------------------------------------
// Kernel 1: feat = relu(target @ W1 + b1).  One wave per 16x16 C tile,
// K stepped by 4 with v_wmma_f32_16x16x4_f32.  2048 waves total.
__global__ __launch_bounds__(256)
void gemm1_relu(const float* __restrict__ target, const float* __restrict__ W1,
                const float* __restrict__ b1, float* __restrict__ feat) {
  const int wave = (blockIdx.x * blockDim.x + threadIdx.x) >> 5;
  const int lane = threadIdx.x & 31;
  const int tm = wave >> 4;          // 0..127  (2048/16 row tiles)
  const int tn = wave & 15;          // 0..15   (256/16  col tiles)
  const int nlo  = lane & 15;
  const int half = lane >> 4;        // 0|1
  const int m = tm * 16 + nlo;       // A-row for this lane
  const int n = tn * 16 + nlo;       // B/C-col for this lane
  const int kb = half << 1;          // lane k-offset 0|2

  v8f acc = {};
  const float* arow = target + (size_t)m * DIN + kb;
  const float* bcol = W1 + (size_t)kb * DB + n;
  for (int k = 0; k < DIN; k += 4) {
    v2f a = *(const v2f*)arow;       // A[m][k+kb], A[m][k+kb+1]
    v2f b;
    b.x = bcol[0];                   // W1[k+kb  ][n]
    b.y = bcol[DB];                  // W1[k+kb+1][n]
    acc = wmma4(a, b, acc);
    arow += 4;
    bcol += 4 * DB;
  }
  const float bias = b1[n];
  const int mbase = tm * 16 + (half << 3);
#pragma unroll
  for (int r = 0; r < 8; ++r) {
    float v = acc[r] + bias;
    feat[(size_t)(mbase + r) * DB + n] = v > 0.0f ? v : 0.0f;
  }
}

// ---------------------------------------------------------------------------
// Kernel 2: S[c] += normalized bank row, counts[c] += 1.
// Bank = [feat(2048) ; hist(63488)].  One wave per row-batch; 32KB LDS
// per-class accumulator, flushed once per block via global atomics.
__global__ __launch_bounds__(256)
void class_sums(const float* __restrict__ feat, const int* __restrict__ curl,
                const float* __restrict__ hist, const int* __restrict__ histl,
                float* __restrict__ S, float* __restrict__ counts) {
  __shared__ float ls[NCLSP * DB];   // 32 KB
  __shared__ float lc[NCLSP];
  for (int i = threadIdx.x; i < NCLSP * DB; i += blockDim.x) ls[i] = 0.0f;
  if (threadIdx.x < NCLSP) lc[threadIdx.x] = 0.0f;
  __syncthreads();

  const int lane = threadIdx.x & 31;
  const int wid  = threadIdx.x >> 5;
  const int gw   = blockIdx.x * 8 + wid;     // 2048 waves
  const int rows_per_wave = NBANK / 2048;    // 32
  const int row0 = gw * rows_per_wave;

  for (int r = 0; r < rows_per_wave; ++r) {
    const int row = row0 + r;
    const float* p;
    int lbl;
    if (row < BROWS) { p = feat + (size_t)row * DB;          lbl = curl[row]; }
    else             { p = hist + (size_t)(row - BROWS) * DB; lbl = histl[row - BROWS]; }
    float x[8], ss = 0.0f;
#pragma unroll
    for (int j = 0; j < 8; ++j) { x[j] = p[lane + 32 * j]; ss += x[j] * x[j]; }
#pragma unroll
    for (int off = 16; off > 0; off >>= 1) ss += __shfl_xor(ss, off, 32);
    const float scale = 1.0f / (sqrtf(ss) + EPSN);
    float* dst = ls + lbl * DB;
#pragma unroll
    for (int j = 0; j < 8; ++j) atomicAdd(&dst[lane + 32 * j], x[j] * scale);
    if (lane == 0) atomicAdd(&lc[lbl], 1.0f);
  }
  __syncthreads();
  for (int i = threadIdx.x; i < NCLSP * DB; i += blockDim.x) atomicAdd(&S[i], ls[i]);
  if (threadIdx.x < NCLSP) atomicAdd(&counts[threadIdx.x], lc[threadIdx.x]);
}

// ---------------------------------------------------------------------------
// Kernel: build ScT[k][c] = S[c][k]/count[c] (0 for empty/pad) and
// WcPad[k][c] = Wc[k][c] (0 for pad col 31).
__global__ void build_mats(const float* __restrict__ S, const float* __restrict__ counts,
                           const float* __restrict__ Wc,
                           float* __restrict__ ScT, float* __restrict__ WcPad) {
  int idx = blockIdx.x * blockDim.x + threadIdx.x;
  if (idx >= DB * NCLSP) return;
  const int k = idx >> 5, c = idx & 31;
  float sc = 0.0f, wc = 0.0f;
  if (c < NCLS) {
    const float cnt = counts[c];
    if (cnt > 0.0f) sc = S[c * DB + k] / cnt;
    wc = Wc[k * NCLS + c];
  }
  ScT[idx] = sc;
  WcPad[idx] = wc;
}

// ---------------------------------------------------------------------------
// Kernel 3: scores = feat @ ScT, logits = feat @ WcPad + bc (via WMMA),
// then per-row argmax/argmin, log-softmax loss, deterministic reduction.
// One wave per 16-row tile; 8 waves (128 rows) per block; 16 blocks.
__global__ __launch_bounds__(256)
void classify(const float* __restrict__ feat, const float* __restrict__ ScT,
              const float* __restrict__ WcPad, const float* __restrict__ counts,
              const float* __restrict__ bc, float* __restrict__ out,
              float* __restrict__ partial) {
  __shared__ float sS[8 * 16 * 32];  // 16 KB scores
  __shared__ float sL[8 * 16 * 32];  // 16 KB logits (pre-bias)
  __shared__ float sred[256];

  const int lane = threadIdx.x & 31;
  const int wid  = threadIdx.x >> 5;
  const int tile = blockIdx.x * 8 + wid;  // 0..127
  const int nlo  = lane & 15;
  const int half = lane >> 4;
  const int m    = tile * 16 + nlo;
  const int kb   = half << 1;

  v8f aS0 = {}, aS1 = {}, aL0 = {}, aL1 = {};
  const float* arow = feat + (size_t)m * DB + kb;
  const float* s0 = ScT + kb * NCLSP + nlo;
  const float* w0 = WcPad + kb * NCLSP + nlo;
  for (int k = 0; k < DB; k += 4) {
    v2f a = *(const v2f*)arow;
    v2f b;
    b.x = s0[0];       b.y = s0[NCLSP];        aS0 = wmma4(a, b, aS0);
    b.x = s0[16];      b.y = s0[NCLSP + 16];   aS1 = wmma4(a, b, aS1);
    b.x = w0[0];       b.y = w0[NCLSP];        aL0 = wmma4(a, b, aL0);
    b.x = w0[16];      b.y = w0[NCLSP + 16];   aL1 = wmma4(a, b, aL1);
    arow += 4;
    s0 += 4 * NCLSP;
    w0 += 4 * NCLSP;
  }

  float* mS = sS + wid * 16 * 32;
  float* mL = sL + wid * 16 * 32;
  const int rbase = half << 3;
#pragma unroll
  for (int r = 0; r < 8; ++r) {
    mS[(rbase + r) * 32 + nlo]      = aS0[r];
    mS[(rbase + r) * 32 + nlo + 16] = aS1[r];
    mL[(rbase + r) * 32 + nlo]      = aL0[r];
    mL[(rbase + r) * 32 + nlo + 16] = aL1[r];
  }
  __syncthreads();

  float lossr = 0.0f;
  if (threadIdx.x < 128) {
    const int lr = threadIdx.x;                 // local row 0..127
    const int grow = blockIdx.x * 128 + lr;     // global row
    const float* rowS = sS + (lr >> 4) * 512 + (lr & 15) * 32;
    const float* rowL = sL + (lr >> 4) * 512 + (lr & 15) * 32;

    // labels_tar: argmin of masked mean cosine distance == argmax score
    float bestS = -3.402823466e+38f;
    int bi = 0;
    for (int c = 0; c < NCLS; ++c) {
      if (counts[c] > 0.0f) {
        const float v = rowS[c];
        if (v > bestS) { bestS = v; bi = c; }
      }
    }
    // logits (+bias), argmax, log-softmax loss at class bi
    float lg[NCLS];
    float lmax = -3.402823466e+38f;
    int pi = 0;
    for (int c = 0; c < NCLS; ++c) {
      lg[c] = rowL[c] + bc[c];
      if (lg[c] > lmax) { lmax = lg[c]; pi = c; }
    }
    float se = 0.0f;
    for (int c = 0; c < NCLS; ++c) se += __expf(lg[c] - lmax);
    const float logZ = lmax + __logf(se);
    lossr = logZ - lg[bi];

    out[grow]         = (float)pi;   // labels_post
    out[BROWS + grow] = (float)bi;   // labels_tar
  }

  // deterministic block loss reduction
  sred[threadIdx.x] = lossr;
  __syncthreads();
  for (int s = 128; s > 0; s >>= 1) {
    if (threadIdx.x < s) sred[threadIdx.x] += sred[threadIdx.x + s];
    __syncthreads();
  }
  if (threadIdx.x == 0) partial[blockIdx.x] = sred[0];
}

// ---------------------------------------------------------------------------
__global__ void write_loss(const float* __restrict__ partial, float* __restrict__ out) {
  float s = 0.0f;
  for (int i = 0; i < 16; ++i) s += partial[i];
  out[2 * BROWS] = s / (float)BROWS;
}

// ---------------------------------------------------------------------------
extern "C" void kernel_launch(void* const* d_in, const int* in_sizes, int n_in,
                              void* d_out, int out_size, void* d_ws, size_t ws_size,
                              hipStream_t stream) {
  const float* target = (const float*)d_in[0];   // [2048, 2048]
  const int*   curl   = (const int*)d_in[1];     // [2048]
  const float* hist   = (const float*)d_in[2];   // [63488, 256]
  const int*   histl  = (const int*)d_in[3];     // [63488]
  const float* W1     = (const float*)d_in[4];   // [2048, 256]
  const float* b1     = (const float*)d_in[5];   // [256]
  const float* Wc     = (const float*)d_in[6];   // [256, 31]
  const float* bc     = (const float*)d_in[7];   // [31]
  float* out = (float*)d_out;                    // [2048 | 2048 | 1]
  float* ws  = (float*)d_ws;

  float* feat = ws + FEAT_OFF;
  float* S    = ws + S_OFF;
  float* cnt  = ws + CNT_OFF;
  float* ScT  = ws + SCT_OFF;
  float* WcP  = ws + WCP_OFF;
  float* part = ws + PART_OFF;

  // zero S + counts (8224 floats)
  zero_kernel<<<(8224 + 255) / 256, 256, 0, stream>>>(S, 8224);
  // feat = relu(target @ W1 + b1): 2048 waves, 8 per block
  gemm1_relu<<<256, 256, 0, stream>>>(target, W1, b1, feat);
  // per-class sums of normalized bank rows
  class_sums<<<256, 256, 0, stream>>>(feat, curl, hist, histl, S, cnt);
  // scaled/transposed class matrix + padded classifier weights
  build_mats<<<(DB * NCLSP + 255) / 256, 256, 0, stream>>>(S, cnt, Wc, ScT, WcP);
  // cluster assignment + classifier + loss partials
  classify<<<16, 256, 0, stream>>>(feat, ScT, WcP, cnt, bc, out, part);
  // final loss
  write_loss<<<1, 1, 0, stream>>>(part, out);
}